// Biaffine_24300924961239
// MI455X (gfx1250) — compile-verified
//
#include <hip/hip_runtime.h>

typedef __attribute__((ext_vector_type(2))) float v2f;
typedef __attribute__((ext_vector_type(4))) float v4f;
typedef __attribute__((ext_vector_type(8))) float v8f;

#define B_ 4
#define S_ 512
#define D_ 512
#define R_ 64

// ---------------------------------------------------------------------------
// K1: hbm[b][j] = mean_y h_backward[b][y][j]   (coalesced column reduction)
// ---------------------------------------------------------------------------
__global__ void k_colmean(const float* __restrict__ hb, float* __restrict__ hbm) {
  int t = blockIdx.x * blockDim.x + threadIdx.x;  // 0 .. B*D-1
  int b = t / D_;
  int j = t % D_;
  const float* p = hb + (size_t)b * S_ * D_ + j;
  float s = 0.f;
  for (int y = 0; y < S_; ++y) s += p[(size_t)y * D_];
  hbm[t] = s * (1.0f / (float)S_);
}

// ---------------------------------------------------------------------------
// K2: cvec[b][r] = bias[r] + sum_j hbm[b][j] * U2[(D+j)*R + r]   (tiny)
// ---------------------------------------------------------------------------
__global__ void k_cvec(const float* __restrict__ hbm, const float* __restrict__ U2,
                       const float* __restrict__ bias, float* __restrict__ cvec) {
  int t = threadIdx.x;         // 256 threads: b = t/64, r = t%64
  int b = t >> 6;
  int r = t & 63;
  float acc = bias[r];
  const float* hbb = hbm + b * D_;
  for (int j = 0; j < D_; ++j)
    acc += hbb[j] * U2[(size_t)(D_ + j) * R_ + r];
  cvec[t] = acc;
}

// ---------------------------------------------------------------------------
// K3: W[b][i][r] = U1[i][r][:] . hbm[b][:] + U2top[i][r]
// Streams the 64 MB U1 tensor once; one wave per (i,r) row, coalesced b128
// loads, hbm staged in LDS, shfl_xor wave32 reduction.
// ---------------------------------------------------------------------------
__global__ void k_foldU1(const float* __restrict__ U1, const float* __restrict__ hbm,
                         const float* __restrict__ U2, float* __restrict__ W) {
  __shared__ float s_hbm[B_ * D_];  // 8 KB
  for (int t = threadIdx.x; t < B_ * D_; t += blockDim.x) s_hbm[t] = hbm[t];
  __syncthreads();

  int lane = threadIdx.x & 31;
  int wave = threadIdx.x >> 5;
  int row  = blockIdx.x * 8 + wave;          // flat (i*R + r), 0..D*R-1
  const float* urow = U1 + (size_t)row * D_; // 2 KB contiguous row

  float acc[B_] = {0.f, 0.f, 0.f, 0.f};
#pragma unroll
  for (int c = 0; c < 4; ++c) {
    int j = c * 128 + lane * 4;
    v4f u = *(const v4f*)(urow + j);
#pragma unroll
    for (int b = 0; b < B_; ++b) {
      v4f h = *(const v4f*)(&s_hbm[b * D_ + j]);
      acc[b] += u.x * h.x + u.y * h.y + u.z * h.z + u.w * h.w;
    }
  }
#pragma unroll
  for (int b = 0; b < B_; ++b) {
#pragma unroll
    for (int off = 16; off > 0; off >>= 1)
      acc[b] += __shfl_xor(acc[b], off, 32);
  }
  if (lane == 0) {
    float u2t = U2[row];  // U2top is [D,R] row-major -> flat index == row
#pragma unroll
    for (int b = 0; b < B_; ++b)
      W[(size_t)b * D_ * R_ + row] = acc[b] + u2t;
  }
}

// ---------------------------------------------------------------------------
// K4: out[b] = hf[b] (SxD) @ W[b] (DxR) + cvec[b]   via V_WMMA_F32_16X16X4_F32
// One wave per 16x16 output tile; 512 tiles total.
// A frag (16x4): lane<16 -> A[lane][k0..k0+1]; lane>=16 -> A[lane-16][k0+2..3]
// B frag (4x16): v0 = B[k0+2*half][n0 + (lane&15)], v1 = next K row
// C/D: N = lane&15, M = v + 8*(lane>>4)
// ---------------------------------------------------------------------------
__global__ void k_gemm_wmma(const float* __restrict__ hf, const float* __restrict__ W,
                            const float* __restrict__ cvec, float* __restrict__ out) {
  int lane = threadIdx.x & 31;
  int wid  = blockIdx.x * (blockDim.x >> 5) + (threadIdx.x >> 5);  // 0..511
  int b    = wid >> 7;          // 128 tiles per batch
  int rem  = wid & 127;
  int m0   = (rem >> 2) << 4;   // 32 m-tiles
  int n0   = (rem & 3) << 4;    // 4 n-tiles
  int half = lane >> 4;
  int lrow = lane & 15;

  const float* Arow = hf + ((size_t)b * S_ + (m0 + lrow)) * D_ + 2 * half;
  const float* Wb   = W + (size_t)b * D_ * R_ + n0 + lrow;

  v8f acc = {};
#pragma unroll 4
  for (int k = 0; k < D_; k += 4) {
    v2f a = *(const v2f*)(Arow + k);       // A[m][k+2h], A[m][k+2h+1]
    int kb = k + 2 * half;
    v2f bf;
    bf.x = Wb[(size_t)kb * R_];            // B[k+2h][n]
    bf.y = Wb[(size_t)(kb + 1) * R_];      // B[k+2h+1][n]
    acc = __builtin_amdgcn_wmma_f32_16x16x4_f32(
        /*neg_a=*/false, a, /*neg_b=*/false, bf,
        /*c_mod=*/(short)0, acc, /*reuse_a=*/false, /*reuse_b=*/false);
  }

  float cb = cvec[b * R_ + n0 + lrow];
  float* orow = out + ((size_t)b * S_ + m0 + 8 * half) * R_ + n0 + lrow;
#pragma unroll
  for (int v = 0; v < 8; ++v)
    orow[(size_t)v * R_] = acc[v] + cb;
}

// ---------------------------------------------------------------------------
extern "C" void kernel_launch(void* const* d_in, const int* in_sizes, int n_in,
                              void* d_out, int out_size, void* d_ws, size_t ws_size,
                              hipStream_t stream) {
  const float* hf   = (const float*)d_in[0];  // [B,S,D]
  const float* hb   = (const float*)d_in[1];  // [B,S,D]
  const float* U1   = (const float*)d_in[2];  // [D,R,D]
  const float* U2   = (const float*)d_in[3];  // [2D,R]
  const float* bias = (const float*)d_in[4];  // [R]
  float* out = (float*)d_out;                 // [B,S,R]

  float* hbm  = (float*)d_ws;        // B*D      = 2048 floats
  float* W    = hbm + B_ * D_;       // B*D*R    = 131072 floats
  float* cvec = W + B_ * D_ * R_;    // B*R      = 256 floats  (~534 KB total)

  k_colmean<<<(B_ * D_) / 256, 256, 0, stream>>>(hb, hbm);
  k_cvec<<<1, 256, 0, stream>>>(hbm, U2, bias, cvec);
  k_foldU1<<<(D_ * R_) / 8, 256, 0, stream>>>(U1, hbm, U2, W);
  k_gemm_wmma<<<64, 256, 0, stream>>>(hf, W, cvec, out);
}